// ParallelHead_77610059039043
// MI455X (gfx1250) — compile-verified
//
#include <hip/hip_runtime.h>
#include <hip/hip_bf16.h>

// ---------------------------------------------------------------------------
// ParallelHead (MEME/Agent57 dueling heads) — fully affine network, so we fold
// all weights into a single [576 x 512] matrix G and one GEMM:
//   out[b, n*A+j] = x[b,:] . G[n*A+j,:] + gbias[n*A+j]
// fp32 throughout via V_WMMA_F32_16X16X4_F32 (exact math, matrix pipes).
// Main GEMM uses double-buffered GLOBAL_LOAD_ASYNC_TO_LDS_B128 staging.
// ---------------------------------------------------------------------------

#define NH  32      // heads
#define AH  18      // actions
#define DH  512     // dim
#define BSZ 4096    // batch
#define NA  (NH*AH) // 576

typedef __attribute__((ext_vector_type(2))) float v2f;
typedef __attribute__((ext_vector_type(8))) float v8f;
typedef __attribute__((ext_vector_type(4))) int   v4i;

#define AS1 __attribute__((address_space(1)))
#define AS3 __attribute__((address_space(3)))

__device__ __forceinline__ v8f wmma_f32(v2f a, v2f b, v8f c) {
  // 8 args: (neg_a, A, neg_b, B, c_mod, C, reuse_a, reuse_b)
  return __builtin_amdgcn_wmma_f32_16x16x4_f32(false, a, false, b,
                                               (short)0, c, false, false);
}

__device__ __forceinline__ v8f zero8() {
  v8f z;
  #pragma unroll
  for (int i = 0; i < 8; ++i) z[i] = 0.f;
  return z;
}

// 16-byte async copy global -> LDS (per-lane, ASYNCcnt-tracked, no VGPR data).
__device__ __forceinline__ void async_cp16(const float* g, float* l) {
#if __has_builtin(__builtin_amdgcn_global_load_async_to_lds_b128)
  __builtin_amdgcn_global_load_async_to_lds_b128(
      (AS1 v4i*)g, (AS3 v4i*)l, 0, 0);
#else
  unsigned loff = (unsigned)(unsigned long long)(AS3 float*)l;
  unsigned long long ga = (unsigned long long)g;
  asm volatile("global_load_async_to_lds_b128 %0, %1, off"
               :: "v"(loff), "v"(ga) : "memory");
#endif
}

template <int N>
__device__ __forceinline__ void wait_async() {
#if __has_builtin(__builtin_amdgcn_s_wait_asynccnt)
  __builtin_amdgcn_s_wait_asynccnt((short)N);
#else
  asm volatile("s_wait_asynccnt %0" :: "i"(N) : "memory");
#endif
}

// ---------------------------------------------------------------------------
// Kernel 1: per head n, T[n] = U[n] @ W2[n]   (U = Wv + Wa - mean_a Wa)
// T stored padded to 32 rows (rows >= 18 are zero).  Also partial bias:
//   g1[n,j] = bv[n] + ba[n,j] - mean(ba[n,:]) + sum_e U[n,j,e]*b2[n,e]
// Row AH of u_lds is a zero row so padded A-frag rows need no EXEC guard.
// ---------------------------------------------------------------------------
__global__ __launch_bounds__(256) void fold_T_kernel(
    const float* __restrict__ W2,   // [N, D, D]
    const float* __restrict__ Wv,   // [N, D]
    const float* __restrict__ Wa,   // [N, A, D]
    const float* __restrict__ b2,   // [N, D]
    const float* __restrict__ bv,   // [N]
    const float* __restrict__ ba,   // [N, A]
    float* __restrict__ T,          // [N, 32, D]
    float* __restrict__ g1)         // [N*A]
{
  __shared__ float u_lds[AH + 1][DH + 4];   // +pad: conflict-free frag reads
  const int n = blockIdx.x;
  const int t = threadIdx.x;

  // Build U[n,j,e] into LDS; zero the spill row AH.
  for (int e = t; e < DH + 4; e += 256) u_lds[AH][e] = 0.f;
  for (int e = t; e < DH; e += 256) {
    float m = 0.f;
    #pragma unroll 6
    for (int j = 0; j < AH; ++j) m += Wa[(size_t)(n*AH + j)*DH + e];
    m *= (1.0f/AH);
    const float wve = Wv[(size_t)n*DH + e];
    #pragma unroll 6
    for (int j = 0; j < AH; ++j)
      u_lds[j][e] = wve + Wa[(size_t)(n*AH + j)*DH + e] - m;
  }
  __syncthreads();

  const int wave = t >> 5, lane = t & 31;
  const int lo = lane & 15, hi = lane >> 4;
  const int d0 = wave * 64;                 // 8 waves cover d = 0..511
  const int j1 = 16 + lo;
  const int r1 = (j1 < AH) ? j1 : AH;       // clamp to zero row (no divergence)

  v8f acc[2][4];
  #pragma unroll
  for (int r = 0; r < 2; ++r)
    #pragma unroll
    for (int c = 0; c < 4; ++c) acc[r][c] = zero8();

  const float* __restrict__ W2n = W2 + (size_t)n * DH * DH;

  for (int e0 = 0; e0 < DH; e0 += 4) {
    const int ea = e0 + 2*hi;
    v2f a0, a1;
    a0.x = u_lds[lo][ea]; a0.y = u_lds[lo][ea + 1];   // rows 0..15
    a1.x = u_lds[r1][ea]; a1.y = u_lds[r1][ea + 1];   // rows 16..31 (0 if pad)
    #pragma unroll
    for (int c = 0; c < 4; ++c) {
      const int d = d0 + c*16 + lo;
      v2f b;
      b.x = W2n[(size_t)ea * DH + d];        // coalesced across lanes
      b.y = W2n[(size_t)(ea + 1) * DH + d];
      acc[0][c] = wmma_f32(a0, b, acc[0][c]);
      acc[1][c] = wmma_f32(a1, b, acc[1][c]);
    }
  }

  float* __restrict__ Tn = T + (size_t)n * 32 * DH;
  #pragma unroll
  for (int rt = 0; rt < 2; ++rt)
    #pragma unroll
    for (int c = 0; c < 4; ++c) {
      const int d = d0 + c*16 + lo;
      #pragma unroll
      for (int r = 0; r < 8; ++r) {
        const int j = rt*16 + r + 8*hi;
        Tn[(size_t)j * DH + d] = acc[rt][c][r];
      }
    }

  if (t < AH) {
    float mba = 0.f;
    #pragma unroll 6
    for (int j = 0; j < AH; ++j) mba += ba[n*AH + j];
    mba *= (1.0f/AH);
    float s = bv[n] + ba[n*AH + t] - mba;
    for (int e = 0; e < DH; ++e) s += u_lds[t][e] * b2[(size_t)n*DH + e];
    g1[n*AH + t] = s;
  }
}

// ---------------------------------------------------------------------------
// Kernel 2: per head n, G[n*A+j, :] = T[n,j,:] @ W1[n]   (row n*D+d of W1)
// and final bias gb[n,j] = g1[n,j] + sum_d T[n,j,d]*b1[n*D+d].
// T (2 MB) is L2-resident; A-frag reads come straight from L2.
// ---------------------------------------------------------------------------
__global__ __launch_bounds__(256) void fold_G_kernel(
    const float* __restrict__ W1,   // [N*D, D]
    const float* __restrict__ b1,   // [N*D]
    const float* __restrict__ T,    // [N, 32, D]
    const float* __restrict__ g1,   // [N*A]
    float* __restrict__ G,          // [NA, D]
    float* __restrict__ gb)         // [NA]
{
  const int n = blockIdx.x;
  const int t = threadIdx.x;
  const int wave = t >> 5, lane = t & 31;
  const int lo = lane & 15, hi = lane >> 4;
  const int k0 = wave * 64;                 // 8 waves cover k = 0..511

  const float* __restrict__ Tn  = T  + (size_t)n * 32 * DH;
  const float* __restrict__ W1n = W1 + (size_t)n * DH * DH;

  v8f acc[2][4];
  #pragma unroll
  for (int r = 0; r < 2; ++r)
    #pragma unroll
    for (int c = 0; c < 4; ++c) acc[r][c] = zero8();

  for (int dd = 0; dd < DH; dd += 4) {
    const int da = dd + 2*hi;
    v2f a0, a1;
    a0.x = Tn[(size_t)lo * DH + da];        a0.y = Tn[(size_t)lo * DH + da + 1];
    a1.x = Tn[(size_t)(16 + lo) * DH + da]; a1.y = Tn[(size_t)(16 + lo) * DH + da + 1];
    #pragma unroll
    for (int c = 0; c < 4; ++c) {
      const int k = k0 + c*16 + lo;
      v2f b;
      b.x = W1n[(size_t)da * DH + k];        // coalesced, read exactly once
      b.y = W1n[(size_t)(da + 1) * DH + k];
      acc[0][c] = wmma_f32(a0, b, acc[0][c]);
      acc[1][c] = wmma_f32(a1, b, acc[1][c]);
    }
  }

  #pragma unroll
  for (int rt = 0; rt < 2; ++rt)
    #pragma unroll
    for (int c = 0; c < 4; ++c) {
      const int k = k0 + c*16 + lo;
      #pragma unroll
      for (int r = 0; r < 8; ++r) {
        const int j = rt*16 + r + 8*hi;
        if (j < AH) G[((size_t)n*AH + j) * DH + k] = acc[rt][c][r];
      }
    }

  if (t < AH) {
    float s = g1[n*AH + t];
    for (int d = 0; d < DH; ++d)
      s += Tn[(size_t)t * DH + d] * b1[(size_t)n*DH + d];
    gb[n*AH + t] = s;
  }
}

// ---------------------------------------------------------------------------
// Kernel 3: out[B, 576] = x[B,512] @ G^T + gb.  Block tile 128x64, 8 waves,
// each wave a 32x32 register tile.  Double-buffered LDS tiles filled with
// GLOBAL_LOAD_ASYNC_TO_LDS_B128 (ASYNCcnt): next tile DMA overlaps WMMA.
// Row pad of +4 floats -> 16B-aligned chunks and conflict-free frag reads.
// ---------------------------------------------------------------------------
__global__ __launch_bounds__(256) void main_gemm_kernel(
    const float* __restrict__ x,    // [B, D]
    const float* __restrict__ G,    // [NA, D]
    const float* __restrict__ gb,   // [NA]
    float* __restrict__ out)        // [B, NA]
{
  __shared__ __align__(16) float xs[2][128][36];   // 36.0 KB
  __shared__ __align__(16) float gs[2][64][36];    // 18.0 KB
  const int t  = threadIdx.x;
  const int bm = blockIdx.x * 128;              // 32 tiles in M
  const int cn = blockIdx.y * 64;               //  9 tiles in N'
  const int wave = t >> 5, lane = t & 31;
  const int lo = lane & 15, hi = lane >> 4;
  const int wm = (wave >> 1) * 32;              // wave row offset in block
  const int wn = (wave & 1) * 32;               // wave col offset in block

  // 6 async 16B chunks per thread per K-chunk (4 for x tile, 2 for G tile).
  auto issue_tile = [&](int buf, int kc) {
    #pragma unroll
    for (int i = 0; i < 4; ++i) {               // x tile: 128x32 floats
      const int idx = t + i*256;
      const int row = idx >> 3, c4 = idx & 7;
      async_cp16(&x[(size_t)(bm + row)*DH + kc + c4*4], &xs[buf][row][c4*4]);
    }
    #pragma unroll
    for (int i = 0; i < 2; ++i) {               // G tile: 64x32 floats
      const int idx = t + i*256;
      const int row = idx >> 3, c4 = idx & 7;
      async_cp16(&G[(size_t)(cn + row)*DH + kc + c4*4], &gs[buf][row][c4*4]);
    }
  };

  v8f acc[2][2];
  #pragma unroll
  for (int i = 0; i < 2; ++i)
    #pragma unroll
    for (int j = 0; j < 2; ++j) acc[i][j] = zero8();

  issue_tile(0, 0);
  for (int it = 0; it < DH/32; ++it) {
    const int buf = it & 1;
    if (it + 1 < DH/32) {
      issue_tile(buf ^ 1, (it + 1) * 32);  // prefetch next tile (other buffer)
      wait_async<6>();                      // only older batch must be done
    } else {
      wait_async<0>();
    }
    __syncthreads();                        // all waves' chunks landed in LDS

    #pragma unroll
    for (int k = 0; k < 32; k += 4) {
      const int ka = k + 2*hi;
      v2f a0, a1, b0, b1;
      a0.x = xs[buf][wm + lo][ka];       a0.y = xs[buf][wm + lo][ka + 1];
      a1.x = xs[buf][wm + 16 + lo][ka];  a1.y = xs[buf][wm + 16 + lo][ka + 1];
      b0.x = gs[buf][wn + lo][ka];       b0.y = gs[buf][wn + lo][ka + 1];
      b1.x = gs[buf][wn + 16 + lo][ka];  b1.y = gs[buf][wn + 16 + lo][ka + 1];
      acc[0][0] = wmma_f32(a0, b0, acc[0][0]);
      acc[0][1] = wmma_f32(a0, b1, acc[0][1]);
      acc[1][0] = wmma_f32(a1, b0, acc[1][0]);
      acc[1][1] = wmma_f32(a1, b1, acc[1][1]);
    }
    __syncthreads();   // nobody may overwrite this buffer until all consumed
  }

  #pragma unroll
  for (int mi = 0; mi < 2; ++mi)
    #pragma unroll
    for (int ni = 0; ni < 2; ++ni) {
      const int c = cn + wn + ni*16 + lo;
      const float bias = gb[c];
      #pragma unroll
      for (int r = 0; r < 8; ++r) {
        const int m = bm + wm + mi*16 + r + 8*hi;
        out[(size_t)m * NA + c] = acc[mi][ni][r] + bias;
      }
    }
}

// ---------------------------------------------------------------------------
// Host launcher.  d_in order: x, W1, b1, W2, b2, Wv, bv, Wa, ba (all fp32).
// Workspace (floats): T[N*32*D] | g1[576] | gb[576] | G[576*512]  ~3.3 MB.
// ---------------------------------------------------------------------------
extern "C" void kernel_launch(void* const* d_in, const int* in_sizes, int n_in,
                              void* d_out, int out_size, void* d_ws, size_t ws_size,
                              hipStream_t stream) {
  const float* x  = (const float*)d_in[0];
  const float* W1 = (const float*)d_in[1];
  const float* b1 = (const float*)d_in[2];
  const float* W2 = (const float*)d_in[3];
  const float* b2 = (const float*)d_in[4];
  const float* Wv = (const float*)d_in[5];
  const float* bv = (const float*)d_in[6];
  const float* Wa = (const float*)d_in[7];
  const float* ba = (const float*)d_in[8];
  float* out = (float*)d_out;

  float* ws = (float*)d_ws;
  float* T  = ws;                               // N*32*D = 524288
  float* g1 = ws + (size_t)NH * 32 * DH;        // 576
  float* gb = g1 + NA;                          // 576
  float* G  = gb + NA;                          // NA*D = 294912

  fold_T_kernel<<<NH, 256, 0, stream>>>(W2, Wv, Wa, b2, bv, ba, T, g1);
  fold_G_kernel<<<NH, 256, 0, stream>>>(W1, b1, T, g1, G, gb);
  main_gemm_kernel<<<dim3(BSZ/128, NA/64), 256, 0, stream>>>(x, G, gb, out);
}